// CrossAttentionInpaintingHead_85633057947836
// MI455X (gfx1250) — compile-verified
//
#include <hip/hip_runtime.h>
#include <math.h>

// ---------------------------------------------------------------------------
// CrossAttentionInpaintingHead forward for MI455X (gfx1250, wave32).
// Dense projections: V_WMMA_F32_16X16X4_F32 (full f32 fidelity),
// unrolled K loops.  Big-GEMM weights staged into LDS with the Tensor Data
// Mover (tensor_load_to_lds + s_wait_tensorcnt).  Irregular attention pieces
// use wave-per-row VALU with wave32 shuffle reductions.
// Workspace requirement: ~55M floats (~220 MB).
// ---------------------------------------------------------------------------

#define NSENS   4760
#define KNN_K   16
#define HIDDEN  64
#define LATENT  1024
#define LPD     128
#define POSD    96
#define FACED   32
#define NHEADS  4
#define HDIM    32
#define NTOK    6
#define NBATCH  32
#define ROWS    (NBATCH * NSENS)      // 152320
#define NBRROWS (NSENS * KNN_K)       // 76160
#define MTPB    8                     // M-tiles per block in LDS-staged GEMMs

typedef __attribute__((ext_vector_type(2))) float    v2f;
typedef __attribute__((ext_vector_type(8))) float    v8f;
typedef __attribute__((ext_vector_type(4))) unsigned u32x4;
typedef __attribute__((ext_vector_type(4))) int      i32x4;
typedef __attribute__((ext_vector_type(8))) int      i32x8;

__device__ __forceinline__ v8f wmma4(v2f a, v2f b, v8f c) {
  // D = A(16x4) * B(4x16) + C(16x16), all f32
  return __builtin_amdgcn_wmma_f32_16x16x4_f32(false, a, false, b, (short)0, c,
                                               false, false);
}

__device__ __forceinline__ float wave_reduce_sum(float v) {
  #pragma unroll
  for (int off = 16; off > 0; off >>= 1) v += __shfl_xor(v, off, 32);
  return v;
}

// ---------------------------------------------------------------------------
// Tensor Data Mover: stage a dense 2D f32 tile (rows x cols, contiguous,
// stride == cols) from global memory into LDS.  One wave issues the D#,
// then all waves wait on TENSORcnt and barrier.
// D# layout per CDNA5 ISA.
// ---------------------------------------------------------------------------
__device__ __forceinline__ void tdm_stage_2d(void* lds_dst, const void* gsrc,
                                             unsigned rows, unsigned cols) {
  if (threadIdx.x == 0) {
    unsigned long long ga = (unsigned long long)(uintptr_t)gsrc;
    unsigned lds = (unsigned)(uintptr_t)lds_dst;   // flat low 32 bits == LDS addr
    u32x4 g0;
    g0[0] = 1u;                                    // count=1, user mode, no gather
    g0[1] = lds;                                   // lds_addr (bytes)
    g0[2] = (unsigned)ga;                          // global_addr[31:0]
    g0[3] = ((unsigned)(ga >> 32) & 0x01FFFFFFu)   // global_addr[56:32]
            | (2u << 30);                          // type = 2 ("image")
    i32x8 g1;
    g1[0] = (int)(0x2u << 16);                     // wg_mask=0, data_size=2 (4B)
    g1[1] = (int)((cols & 0xFFFFu) << 16);         // tensor_dim0[15:0] @63:48
    g1[2] = (int)(((cols >> 16) & 0xFFFFu)         // tensor_dim0[31:16]
                | ((rows & 0xFFFFu) << 16));       // tensor_dim1[15:0]
    g1[3] = (int)(((rows >> 16) & 0xFFFFu)         // tensor_dim1[31:16]
                | ((cols & 0xFFFFu) << 16));       // tile_dim0
    g1[4] = (int)(rows & 0xFFFFu);                 // tile_dim1 (tile_dim2 = 0)
    g1[5] = (int)cols;                             // tensor_dim0_stride[31:0]
    g1[6] = 0;                                     // stride hi / dim1_stride lo
    g1[7] = 0;
#if __has_builtin(__builtin_amdgcn_tensor_load_to_lds)
    i32x4 z4 = {};
  #if defined(__clang_major__) && (__clang_major__ >= 23)
    i32x8 z8 = {};
    __builtin_amdgcn_tensor_load_to_lds(g0, g1, z4, z4, z8, 0);
  #else
    __builtin_amdgcn_tensor_load_to_lds(g0, g1, z4, z4, 0);
  #endif
#else
    asm volatile("tensor_load_to_lds %0, %1" :: "s"(g0), "s"(g1) : "memory");
#endif
  }
  __builtin_amdgcn_s_wait_tensorcnt(0);
  __syncthreads();
}

// ---------------------------------------------------------------------------
// biasL[t][j] = b_lat[j] + b_lf[j] + sum_c face_emb[tf[t]][c] * W_lf[c][j]
// ---------------------------------------------------------------------------
__global__ void lf_bias_kernel(const float* __restrict__ face_emb,
                               const int* __restrict__ tf,
                               const float* __restrict__ W_lf,
                               const float* __restrict__ b_lf,
                               const float* __restrict__ b_lat,
                               float* __restrict__ biasL) {
  int t = blockIdx.x;      // 0..5
  int j = threadIdx.x;     // 0..127
  const float* fe = face_emb + tf[t] * FACED;
  float s = b_lat[j] + b_lf[j];
  #pragma unroll
  for (int c = 0; c < FACED; ++c) s = fmaf(fe[c], W_lf[c * LPD + j], s);
  biasL[t * LPD + j] = s;
}

// ---------------------------------------------------------------------------
// latent_kv[192,128] = latent_seq[192,1024] @ W_lat + biasL[row%6][:]
// ---------------------------------------------------------------------------
__global__ void gemm_latent_kv(const float* __restrict__ A,
                               const float* __restrict__ W_lat,
                               const float* __restrict__ biasL,
                               float* __restrict__ C) {
  const int N = LPD, K = LATENT;
  int wid = (int)((blockIdx.x * blockDim.x + threadIdx.x) >> 5);
  int tiles_n = N >> 4;
  if (wid >= ((NBATCH * NTOK) >> 4) * tiles_n) return;
  int mt = wid / tiles_n, nt = wid - mt * tiles_n;
  int lane = threadIdx.x & 31, half = lane >> 4, l16 = lane & 15;
  const float* ap = A + (size_t)(mt * 16 + l16) * K;
  int ncol = nt * 16 + l16;
  v8f acc = {};
  #pragma unroll 8
  for (int kk = 0; kk < K; kk += 4) {
    int ka = kk + half * 2;
    v2f a; a.x = ap[ka]; a.y = ap[ka + 1];
    v2f b; b.x = W_lat[ka * N + ncol]; b.y = W_lat[(ka + 1) * N + ncol];
    acc = wmma4(a, b, acc);
  }
  #pragma unroll
  for (int r = 0; r < 8; ++r) {
    int row = mt * 16 + (half << 3) + r;
    C[row * N + ncol] = acc[r] + biasL[(row % NTOK) * LPD + ncol];
  }
}

// ---------------------------------------------------------------------------
// LDS-staged WMMA GEMM for N=128, K=128, M % 16 == 0:
//   C[M,128] = A[M,128] @ B[128,128] + bias
// Block = 8 waves; wave w owns N-tile w; block sweeps MTPB M-tiles.
// B staged once per block via TDM; fragments read from LDS.
// ---------------------------------------------------------------------------
__global__ void gemm_nk128_ldsb(const float* __restrict__ A,
                                const float* __restrict__ B,
                                const float* __restrict__ bias,
                                float* __restrict__ C, int M) {
  __shared__ float Bs[LPD * LPD];                      // 64 KB
  tdm_stage_2d(Bs, B, LPD, LPD);

  int nt = (int)(threadIdx.x >> 5);                    // wave -> N-tile 0..7
  int lane = threadIdx.x & 31, half = lane >> 4, l16 = lane & 15;
  int ncol = nt * 16 + l16;
  float bv = bias[ncol];
  for (int mi = 0; mi < MTPB; ++mi) {
    int mt = blockIdx.x * MTPB + mi;
    if (mt * 16 >= M) break;                           // wave-uniform
    const float* ap = A + (size_t)(mt * 16 + l16) * LPD;
    v8f acc = {};
    #pragma unroll
    for (int kk = 0; kk < LPD; kk += 4) {
      int ka = kk + half * 2;
      v2f a; a.x = ap[ka]; a.y = ap[ka + 1];
      v2f b; b.x = Bs[ka * LPD + ncol]; b.y = Bs[(ka + 1) * LPD + ncol];
      acc = wmma4(a, b, acc);
    }
    size_t base = (size_t)(mt * 16 + (half << 3)) * LPD + ncol;
    #pragma unroll
    for (int r = 0; r < 8; ++r) C[base + (size_t)r * LPD] = acc[r] + bv;
  }
}

// ---------------------------------------------------------------------------
// Gathered-A WMMA GEMM over the 128-dim "query" features:
//   A[r][c] = c<96 ? pos_embed[s][c] : face_emb[face_ids[s]][c-96]
//   s = knn ? knn[r] : r            (knn flat (4760*16) matches row index)
// C[M,N] = A @ B + bias.  K = 128 fixed.
// ---------------------------------------------------------------------------
__global__ void gemm_gather_wmma(const float* __restrict__ pos_embed,
                                 const float* __restrict__ face_emb,
                                 const int* __restrict__ face_ids,
                                 const int* __restrict__ knn,
                                 const float* __restrict__ B,
                                 const float* __restrict__ bias,
                                 float* __restrict__ C, int M, int N) {
  const int K = POSD + FACED;  // 128
  int wid = (int)((blockIdx.x * blockDim.x + threadIdx.x) >> 5);
  int tiles_n = N >> 4;
  int total = ((M + 15) >> 4) * tiles_n;
  if (wid >= total) return;
  int mt = wid / tiles_n, nt = wid - mt * tiles_n;
  int lane = threadIdx.x & 31, half = lane >> 4, l16 = lane & 15;
  int mrow = mt * 16 + l16;
  int arow = (mrow < M) ? mrow : 0;
  int sensor = knn ? knn[arow] : arow;
  const float* pp = pos_embed + (size_t)sensor * POSD;
  const float* fp = face_emb + (size_t)face_ids[sensor] * FACED;
  int ncol = nt * 16 + l16;
  v8f acc = {};
  #pragma unroll
  for (int kk = 0; kk < K; kk += 4) {
    int ka = kk + half * 2;             // boundary 96 is a multiple of 4 ->
    v2f a;                              // branch is wave-uniform
    if (ka < POSD) { a.x = pp[ka];        a.y = pp[ka + 1]; }
    else           { a.x = fp[ka - POSD]; a.y = fp[ka - POSD + 1]; }
    v2f b; b.x = B[ka * N + ncol]; b.y = B[(ka + 1) * N + ncol];
    acc = wmma4(a, b, acc);
  }
  float bv = bias[ncol];
  #pragma unroll
  for (int r = 0; r < 8; ++r) {
    int row = mt * 16 + (half << 3) + r;
    if (row < M) C[(size_t)row * N + ncol] = acc[r] + bv;
  }
}

// ---------------------------------------------------------------------------
// Local KNN attention.  One wave per (b,n); lane owns hidden pair (2h,2h+1).
// ---------------------------------------------------------------------------
__global__ void local_attn_kernel(const float* __restrict__ x_flat,
                                  const int* __restrict__ enc_mask,
                                  const int* __restrict__ knn,
                                  const float* __restrict__ q_local,
                                  const float* __restrict__ nbr_static,
                                  const float* __restrict__ W_nbr,
                                  float* __restrict__ local_feat) {
  int wid = (int)((blockIdx.x * blockDim.x + threadIdx.x) >> 5);
  if (wid >= ROWS) return;
  int b = wid / NSENS, n = wid - b * NSENS;
  int lane = threadIdx.x & 31;
  int h0 = lane * 2;
  float q0 = q_local[n * HIDDEN + h0];
  float q1 = q_local[n * HIDDEN + h0 + 1];
  float w00 = W_nbr[h0],          w01 = W_nbr[h0 + 1];           // row 0 of W_nbr[:2]
  float w10 = W_nbr[HIDDEN + h0], w11 = W_nbr[HIDDEN + h0 + 1];  // row 1
  float f0[KNN_K], f1[KNN_K], lg[KNN_K];
  #pragma unroll
  for (int k = 0; k < KNN_K; ++k) {
    int idx = knn[n * KNN_K + k];
    const float* xp = x_flat + ((size_t)b * NSENS + idx) * 2;
    float x0 = xp[0], x1 = xp[1];
    const float* sp = nbr_static + ((size_t)n * KNN_K + k) * HIDDEN + h0;
    f0[k] = fmaf(x0, w00, fmaf(x1, w10, sp[0]));
    f1[k] = fmaf(x0, w01, fmaf(x1, w11, sp[1]));
    float p = wave_reduce_sum(f0[k] * q0 + f1[k] * q1);
    lg[k] = p * 0.125f;                               // 1/sqrt(64)
    if (enc_mask[(size_t)b * NSENS + idx] != 0) lg[k] = -10000.0f;
  }
  float m = lg[0];
  #pragma unroll
  for (int k = 1; k < KNN_K; ++k) m = fmaxf(m, lg[k]);
  float s = 0.0f;
  #pragma unroll
  for (int k = 0; k < KNN_K; ++k) { lg[k] = __expf(lg[k] - m); s += lg[k]; }
  float inv = 1.0f / s;
  float o0 = 0.0f, o1 = 0.0f;
  #pragma unroll
  for (int k = 0; k < KNN_K; ++k) {
    float w = lg[k] * inv;
    o0 = fmaf(w, f0[k], o0);
    o1 = fmaf(w, f1[k], o1);
  }
  float* op = local_feat + (size_t)wid * HIDDEN + h0;
  op[0] = o0; op[1] = o1;
}

// ---------------------------------------------------------------------------
// Global 4-head / 6-token attention.  One wave per (b,n); lane = head dim.
// ---------------------------------------------------------------------------
__global__ void global_attn_kernel(const float* __restrict__ q_g,
                                   const float* __restrict__ k_g,
                                   const float* __restrict__ v_g,
                                   float* __restrict__ global_attn) {
  int wid = (int)((blockIdx.x * blockDim.x + threadIdx.x) >> 5);
  if (wid >= ROWS) return;
  int b = wid / NSENS, n = wid - b * NSENS;
  int lane = threadIdx.x & 31;
  const float scale = 0.1767766952966369f;            // 1/sqrt(32)
  float* op = global_attn + (size_t)wid * LPD;
  #pragma unroll
  for (int h = 0; h < NHEADS; ++h) {
    float qd = q_g[n * LPD + h * HDIM + lane];
    float lg[NTOK];
    #pragma unroll
    for (int t = 0; t < NTOK; ++t) {
      float kd = k_g[(b * NTOK + t) * LPD + h * HDIM + lane];
      lg[t] = wave_reduce_sum(qd * kd) * scale;
    }
    float m = lg[0];
    #pragma unroll
    for (int t = 1; t < NTOK; ++t) m = fmaxf(m, lg[t]);
    float s = 0.0f;
    #pragma unroll
    for (int t = 0; t < NTOK; ++t) { lg[t] = __expf(lg[t] - m); s += lg[t]; }
    float inv = 1.0f / s, o = 0.0f;
    #pragma unroll
    for (int t = 0; t < NTOK; ++t)
      o = fmaf(lg[t] * inv, v_g[(b * NTOK + t) * LPD + h * HDIM + lane], o);
    op[h * HDIM + lane] = o;
  }
}

// ---------------------------------------------------------------------------
// LayerNorm row statistics over comb = [local_feat(64) | gf_o(128)].
// stats[r] = {mu, 1/sqrt(var+1e-5)}.  One wave per row.
// ---------------------------------------------------------------------------
__global__ void stats_kernel(const float* __restrict__ lf,
                             const float* __restrict__ gfo,
                             float* __restrict__ stats) {
  int wid = (int)((blockIdx.x * blockDim.x + threadIdx.x) >> 5);
  if (wid >= ROWS) return;
  int lane = threadIdx.x & 31;
  float s = 0.0f, s2 = 0.0f;
  #pragma unroll
  for (int e = 0; e < 6; ++e) {
    int c = lane + e * 32;
    float v = (c < HIDDEN) ? lf[(size_t)wid * HIDDEN + c]
                           : gfo[(size_t)wid * LPD + c - HIDDEN];
    s += v; s2 = fmaf(v, v, s2);
  }
  s = wave_reduce_sum(s);
  s2 = wave_reduce_sum(s2);
  if (lane == 0) {
    float mu = s * (1.0f / 192.0f);
    float var = s2 * (1.0f / 192.0f) - mu * mu;
    stats[(size_t)wid * 2 + 0] = mu;
    stats[(size_t)wid * 2 + 1] = rsqrtf(var + 1e-5f);
  }
}

// ---------------------------------------------------------------------------
// Fused: LayerNorm -> W_m1 (WMMA, K=192, gathered+normalized A, W_m1 staged
// in LDS via TDM) -> GELU -> W_m2(64->2) -> mask -> out.
// Block = 4 waves = one 16-row tile; block sweeps MTPB M-tiles.
// ---------------------------------------------------------------------------
__global__ void fused_out_kernel(const float* __restrict__ lf,
                                 const float* __restrict__ gfo,
                                 const float* __restrict__ stats,
                                 const float* __restrict__ ln_g,
                                 const float* __restrict__ ln_b,
                                 const float* __restrict__ W_m1,
                                 const float* __restrict__ b_m1,
                                 const float* __restrict__ W_m2,
                                 const float* __restrict__ b_m2,
                                 const int* __restrict__ mask,
                                 float* __restrict__ out) {
  __shared__ float Ws[192 * HIDDEN];                  // 48 KB
  __shared__ float hbuf[16][HIDDEN];                  // 4 KB
  tdm_stage_2d(Ws, W_m1, 192, HIDDEN);

  int w = (int)(threadIdx.x >> 5);                    // wave -> cols 16w..16w+15
  int lane = threadIdx.x & 31, half = lane >> 4, l16 = lane & 15;
  int ncol = w * 16 + l16;
  float bm = b_m1[ncol];
  for (int mi = 0; mi < MTPB; ++mi) {
    int mt = blockIdx.x * MTPB + mi;                  // M divisible by 16*MTPB
    int mrow = mt * 16 + l16;
    float mu = stats[mrow * 2 + 0];
    float rstd = stats[mrow * 2 + 1];
    v8f acc = {};
    #pragma unroll
    for (int kk = 0; kk < 192; kk += 4) {
      int ka = kk + half * 2;                         // boundary 64 % 4 == 0
      v2f a;
      if (ka < HIDDEN) { const float* p = lf + (size_t)mrow * HIDDEN + ka;
                         a.x = p[0]; a.y = p[1]; }
      else             { const float* p = gfo + (size_t)mrow * LPD + ka - HIDDEN;
                         a.x = p[0]; a.y = p[1]; }
      a.x = fmaf((a.x - mu) * rstd, ln_g[ka],     ln_b[ka]);
      a.y = fmaf((a.y - mu) * rstd, ln_g[ka + 1], ln_b[ka + 1]);
      v2f b; b.x = Ws[ka * HIDDEN + ncol]; b.y = Ws[(ka + 1) * HIDDEN + ncol];
      acc = wmma4(a, b, acc);
    }
    __syncthreads();                                  // prev epilogue done
    #pragma unroll
    for (int r = 0; r < 8; ++r) {
      float x = acc[r] + bm;
      // exact GELU: 0.5*x*(1+erf(x/sqrt(2)))
      hbuf[(half << 3) + r][ncol] = 0.5f * x * (1.0f + erff(x * 0.70710678118654752f));
    }
    __syncthreads();
    if (threadIdx.x < 32) {
      int row = (int)threadIdx.x >> 1, ch = (int)threadIdx.x & 1;
      float s = b_m2[ch];
      #pragma unroll
      for (int j = 0; j < HIDDEN; ++j) s = fmaf(hbuf[row][j], W_m2[j * 2 + ch], s);
      int gr = mt * 16 + row;                         // gr = b*NSENS + n
      out[gr * 2 + ch] = s * (float)mask[gr];
    }
  }
}

// ---------------------------------------------------------------------------
static inline int gather_blocks(int M, int N) {
  int waves = ((M + 15) / 16) * (N / 16);
  return (waves + 7) / 8;                    // 256-thread blocks = 8 waves
}

extern "C" void kernel_launch(void* const* d_in, const int* in_sizes, int n_in,
                              void* d_out, int out_size, void* d_ws, size_t ws_size,
                              hipStream_t stream) {
  const float* x_flat     = (const float*)d_in[0];
  const float* latent_seq = (const float*)d_in[1];
  const int*   mask       = (const int*)d_in[2];
  const int*   enc_mask   = (const int*)d_in[3];
  const float* pos_embed  = (const float*)d_in[4];
  const int*   knn        = (const int*)d_in[5];
  const int*   face_ids   = (const int*)d_in[6];
  const int*   tf_map     = (const int*)d_in[7];
  const float* face_emb   = (const float*)d_in[8];
  const float* W_nbr      = (const float*)d_in[9];
  const float* b_nbr      = (const float*)d_in[10];
  const float* W_ql       = (const float*)d_in[11];
  const float* b_ql       = (const float*)d_in[12];
  const float* W_lat      = (const float*)d_in[13];
  const float* b_lat      = (const float*)d_in[14];
  const float* W_lf       = (const float*)d_in[15];
  const float* b_lf       = (const float*)d_in[16];
  const float* W_qg       = (const float*)d_in[17];
  const float* b_qg       = (const float*)d_in[18];
  const float* W_k        = (const float*)d_in[19];
  const float* b_k        = (const float*)d_in[20];
  const float* W_v        = (const float*)d_in[21];
  const float* b_v        = (const float*)d_in[22];
  const float* W_go       = (const float*)d_in[23];
  const float* b_go       = (const float*)d_in[24];
  const float* ln_g       = (const float*)d_in[25];
  const float* ln_b       = (const float*)d_in[26];
  const float* W_m1       = (const float*)d_in[27];
  const float* b_m1       = (const float*)d_in[28];
  const float* W_m2       = (const float*)d_in[29];
  const float* b_m2       = (const float*)d_in[30];
  float* out = (float*)d_out;

  // ---- workspace layout (floats) --------------------------------------
  float* ws = (float*)d_ws;
  float* q_local     = ws;                               // 4760*64
  float* q_g         = q_local + (size_t)NSENS * HIDDEN; // 4760*128
  float* nbr_static  = q_g + (size_t)NSENS * LPD;        // 76160*64
  float* biasL       = nbr_static + (size_t)NBRROWS * HIDDEN;   // 6*128
  float* latent_kv   = biasL + NTOK * LPD;               // 192*128
  float* k_g         = latent_kv + (size_t)NBATCH * NTOK * LPD;
  float* v_g         = k_g + (size_t)NBATCH * NTOK * LPD;
  float* local_feat  = v_g + (size_t)NBATCH * NTOK * LPD;       // 152320*64
  float* global_attn = local_feat + (size_t)ROWS * HIDDEN;      // 152320*128
  float* gf_o        = global_attn + (size_t)ROWS * LPD;        // 152320*128
  float* stats       = gf_o + (size_t)ROWS * LPD;               // 152320*2

  // 1. latent face bias (b_lat + face_emb@W_lf + b_lf)
  lf_bias_kernel<<<NTOK, LPD, 0, stream>>>(face_emb, tf_map, W_lf, b_lf, b_lat, biasL);

  // 2. latent_kv = latent_seq @ W_lat + biasL[t]     (192x128, K=1024)
  gemm_latent_kv<<<12, 256, 0, stream>>>(latent_seq, W_lat, biasL, latent_kv);

  // 3/4. k_g, v_g = latent_kv @ {W_k,W_v} + bias     (192x128, K=128, TDM+LDS B)
  gemm_nk128_ldsb<<<(192 / 16 + MTPB - 1) / MTPB, 256, 0, stream>>>(
      latent_kv, W_k, b_k, k_g, 192);
  gemm_nk128_ldsb<<<(192 / 16 + MTPB - 1) / MTPB, 256, 0, stream>>>(
      latent_kv, W_v, b_v, v_g, 192);

  // 5. q_local = query @ W_ql + b_ql   (4760x64, gathered A)
  gemm_gather_wmma<<<gather_blocks(NSENS, HIDDEN), 256, 0, stream>>>(
      pos_embed, face_emb, face_ids, nullptr, W_ql, b_ql, q_local, NSENS, HIDDEN);

  // 6. q_g = query @ W_qg + b_qg       (4760x128, gathered A)
  gemm_gather_wmma<<<gather_blocks(NSENS, LPD), 256, 0, stream>>>(
      pos_embed, face_emb, face_ids, nullptr, W_qg, b_qg, q_g, NSENS, LPD);

  // 7. nbr_static = gather(query[knn]) @ W_nbr[2:] + b_nbr (76160x64)
  gemm_gather_wmma<<<gather_blocks(NBRROWS, HIDDEN), 256, 0, stream>>>(
      pos_embed, face_emb, face_ids, knn, W_nbr + 2 * HIDDEN, b_nbr,
      nbr_static, NBRROWS, HIDDEN);

  // 8. local KNN attention -> local_feat
  local_attn_kernel<<<(ROWS + 7) / 8, 256, 0, stream>>>(
      x_flat, enc_mask, knn, q_local, nbr_static, W_nbr, local_feat);

  // 9. global token attention -> global_attn
  global_attn_kernel<<<(ROWS + 7) / 8, 256, 0, stream>>>(
      q_g, k_g, v_g, global_attn);

  // 10. gf_o = global_attn @ W_go + b_go  (152320x128, K=128, TDM+LDS B)
  gemm_nk128_ldsb<<<(ROWS / 16) / MTPB, 256, 0, stream>>>(
      global_attn, W_go, b_go, gf_o, ROWS);

  // 11. LayerNorm row stats
  stats_kernel<<<(ROWS + 7) / 8, 256, 0, stream>>>(local_feat, gf_o, stats);

  // 12. fused LN -> W_m1 (TDM+LDS) -> GELU -> W_m2 -> mask -> out
  fused_out_kernel<<<(ROWS / 16) / MTPB, 128, 0, stream>>>(
      local_feat, gf_o, stats, ln_g, ln_b, W_m1, b_m1, W_m2, b_m2, mask, out);
}